// Reblur_Net__67946382622933
// MI455X (gfx1250) — compile-verified
//
#include <hip/hip_runtime.h>
#include <hip/hip_bf16.h>

typedef __attribute__((ext_vector_type(2))) float v2f;
typedef __attribute__((ext_vector_type(8))) float v8f;

#define HH 512
#define WW 512
#define HW (HH * WW)   // 2^18

// out = 0.25f * sharp
__global__ __launch_bounds__(256) void reblur_init(const float* __restrict__ s,
                                                   float* __restrict__ out, int n) {
    int i = blockIdx.x * 256 + threadIdx.x;
    if (i < n) out[i] = 0.25f * s[i];
}

// One DCNv2 pass: nxt = dcn(cur, motion, wmat, bias); acc += 0.25*nxt
// Each wave handles 32 pixels: builds val[32][28] in LDS, then
// D[16x16] = W[16x28(pad)] x val^T via 7 chunks of v_wmma_f32_16x16x4_f32, twice (2 N-tiles).
__global__ __launch_bounds__(256) void dcn_pass(const float* __restrict__ cur,    // [B,3,H,W]
                                                const float* __restrict__ motion, // [B,27,H,W]
                                                const float* __restrict__ wmat,   // [3][27]
                                                const float* __restrict__ bias,   // [3]
                                                float* __restrict__ nxt,          // [B,3,H,W]
                                                float* __restrict__ acc)          // [B,3,H,W]
{
    __shared__ float smem[8][32][28];   // 28 KiB: per-wave 32 pixels x K=28 (27 + pad)

    const int tid  = threadIdx.x;
    const int wave = tid >> 5;
    const int lane = tid & 31;
    const int pixBase = blockIdx.x * 256 + wave * 32;   // linear pixel over B*H*W
    const int pid = pixBase + lane;
    const int b   = pid >> 18;          // HW == 2^18
    const int hw  = pid & (HW - 1);
    const int y   = hw >> 9;
    const int x   = hw & (WW - 1);

    const float* curB = cur    + b * 3  * HW;
    const float* motB = motion + b * 27 * HW;

    // ---- Phase 1: modulated bilinear gather -> val[c*9+t], write to LDS ----
    float val[27];
#pragma unroll
    for (int t = 0; t < 9; ++t) {
        const float dy = motB[(2 * t    ) * HW + hw];
        const float dx = motB[(2 * t + 1) * HW + hw];
        const float m  = motB[(18 + t   ) * HW + hw];
        const float py = (float)(y + (t / 3) - 1) + dy;
        const float px = (float)(x + (t % 3) - 1) + dx;
        const float y0f = floorf(py);
        const float x0f = floorf(px);
        const float wy = py - y0f;
        const float wx = px - x0f;
        const int y0 = (int)y0f;
        const int x0 = (int)x0f;
        const int y1 = y0 + 1;
        const int x1 = x0 + 1;
        const float vy0 = (y0 >= 0 && y0 < HH) ? 1.f : 0.f;
        const float vy1 = (y1 >= 0 && y1 < HH) ? 1.f : 0.f;
        const float vx0 = (x0 >= 0 && x0 < WW) ? 1.f : 0.f;
        const float vx1 = (x1 >= 0 && x1 < WW) ? 1.f : 0.f;
        const int y0c = min(max(y0, 0), HH - 1);
        const int y1c = min(max(y1, 0), HH - 1);
        const int x0c = min(max(x0, 0), WW - 1);
        const int x1c = min(max(x1, 0), WW - 1);
        const float w00 = (1.f - wy) * (1.f - wx) * m * vy0 * vx0;
        const float w01 = (1.f - wy) * wx         * m * vy0 * vx1;
        const float w10 = wy         * (1.f - wx) * m * vy1 * vx0;
        const float w11 = wy         * wx         * m * vy1 * vx1;
        const int i00 = y0c * WW + x0c;
        const int i01 = y0c * WW + x1c;
        const int i10 = y1c * WW + x0c;
        const int i11 = y1c * WW + x1c;
#pragma unroll
        for (int c = 0; c < 3; ++c) {
            const float* ch = curB + c * HW;
            val[c * 9 + t] = w00 * ch[i00] + w01 * ch[i01] + w10 * ch[i10] + w11 * ch[i11];
        }
    }
#pragma unroll
    for (int k = 0; k < 27; ++k) smem[wave][lane][k] = val[k];
    smem[wave][lane][27] = 0.f;   // K pad

    // ---- Phase 2: A fragments (weight, zero-padded 16x28), 7 chunks of K=4 ----
    // A 16x4 f32 layout: lanes 0-15 -> M=lane, VGPR0/1 = K {h,h+1}, h = (lane>>4)*2
    const int am  = lane & 15;
    const int akh = (lane >> 4) * 2;
    float a0[7], a1[7];
#pragma unroll
    for (int kk = 0; kk < 7; ++kk) {
        const int k0 = kk * 4 + akh;
        a0[kk] = (am < 3 && k0     < 27) ? wmat[am * 27 + k0    ] : 0.f;
        a1[kk] = (am < 3 && k0 + 1 < 27) ? wmat[am * 27 + k0 + 1] : 0.f;
    }

    __syncthreads();

    // ---- Phase 3: WMMA. B 4x16 layout: lane n(0-15) holds col N=n, K {h,h+1} ----
    const int row0 = lane & 15;          // N within tile
    const int colk = (lane >> 4) * 2;    // K half
    v8f d0 = {}, d1 = {};
#pragma unroll
    for (int kk = 0; kk < 7; ++kk) {
        v2f a = {a0[kk], a1[kk]};
        const float* bp0 = &smem[wave][row0     ][kk * 4 + colk];
        const float* bp1 = &smem[wave][row0 + 16][kk * 4 + colk];
        v2f bv0 = {bp0[0], bp0[1]};
        v2f bv1 = {bp1[0], bp1[1]};
        d0 = __builtin_amdgcn_wmma_f32_16x16x4_f32(false, a, false, bv0, (short)0, d0, false, false);
        d1 = __builtin_amdgcn_wmma_f32_16x16x4_f32(false, a, false, bv1, (short)0, d1, false, false);
    }

    // ---- Phase 4: store. D layout: lanes 0-15, VGPR r = channel r (rows 3..15 are zero) ----
    if (lane < 16) {
        const float b0 = bias[0], b1 = bias[1], b2 = bias[2];
#pragma unroll
        for (int tile = 0; tile < 2; ++tile) {
            const int p   = pixBase + tile * 16 + lane;
            const int bb  = p >> 18;
            const int phw = p & (HW - 1);
            const float o0 = (tile ? d1[0] : d0[0]) + b0;
            const float o1 = (tile ? d1[1] : d0[1]) + b1;
            const float o2 = (tile ? d1[2] : d0[2]) + b2;
            const int i0 = (bb * 3 + 0) * HW + phw;
            const int i1 = (bb * 3 + 1) * HW + phw;
            const int i2 = (bb * 3 + 2) * HW + phw;
            nxt[i0] = o0;  acc[i0] += 0.25f * o0;
            nxt[i1] = o1;  acc[i1] += 0.25f * o1;
            nxt[i2] = o2;  acc[i2] += 0.25f * o2;
        }
    }
}

extern "C" void kernel_launch(void* const* d_in, const int* in_sizes, int n_in,
                              void* d_out, int out_size, void* d_ws, size_t ws_size,
                              hipStream_t stream) {
    const float* sharp  = (const float*)d_in[0];   // [B,3,512,512]
    const float* motion = (const float*)d_in[1];   // [B,27,512,512]
    const float* wts    = (const float*)d_in[2];   // [3,3,3,3,3] -> 3 x 81
    const float* bias   = (const float*)d_in[3];   // [3,3]
    float* out = (float*)d_out;

    const int n    = in_sizes[0];   // B*3*H*W
    const int npix = n / 3;         // B*H*W (multiple of 256)

    float* buf0 = (float*)d_ws;
    float* buf1 = buf0 + n;

    reblur_init<<<(n + 255) / 256, 256, 0, stream>>>(sharp, out, n);

    const float* cur = sharp;
    float* nxt = buf0;
    for (int i = 0; i < 3; ++i) {
        dcn_pass<<<npix / 256, 256, 0, stream>>>(cur, motion, wts + i * 81, bias + i * 3, nxt, out);
        cur = nxt;
        nxt = (nxt == buf0) ? buf1 : buf0;
    }
}